// Sampler_30296699306294
// MI455X (gfx1250) — compile-verified
//
#include <hip/hip_runtime.h>
#include <math.h>

#define TPB   512    // 16 wave32 waves per block, one block per batch row
#define CAP   2048   // LDS candidate buffer (value+index = 16 KB)
#define KMAX  64     // top_ks < 64 always -> only top-64 can ever be kept
#define NBUF  4      // async double-buffer depth (4 x 8 KB LDS)

// ---- gfx1250 async global->LDS path (ASYNCcnt), with safe fallback --------
#if defined(__AMDGCN__) && __has_builtin(__builtin_amdgcn_global_load_async_to_lds_b128)
  #define HAVE_ASYNC_COPY 1
  typedef int v4i_t __attribute__((vector_size(16)));
  typedef __attribute__((address_space(1))) v4i_t* gv4i_p;   // global int4*
  typedef __attribute__((address_space(3))) v4i_t* lv4i_p;   // LDS int4*
  #if __has_builtin(__builtin_amdgcn_s_wait_asynccnt)
    #define WAIT_ASYNC(n) __builtin_amdgcn_s_wait_asynccnt(n)
  #else
    #define WAIT_ASYNC(n) asm volatile("s_wait_asynccnt %0" :: "i"(n) : "memory")
  #endif
  #define COMPILER_MEM_BARRIER() asm volatile("" ::: "memory")
#else
  #define HAVE_ASYNC_COPY 0
#endif

__global__ __launch_bounds__(TPB) void sampler_topk_topp_kernel(
    const float* __restrict__ logits,
    const float* __restrict__ temps,
    const float* __restrict__ top_ps,
    const float* __restrict__ top_ks,
    const float* __restrict__ noise,
    int*   __restrict__ tok_out,
    float* __restrict__ scaled_out,
    int V)
{
    __shared__ float4   s_abuf[NBUF][TPB];   // async staging: 32 KB
    __shared__ float    s_red[TPB];
    __shared__ float    s_cv[CAP];
    __shared__ int      s_ci[CAP];
    __shared__ float    s_sv[KMAX];
    __shared__ int      s_si[KMAX];
    __shared__ float    s_m;                 // row max of scaled
    __shared__ float    s_q;                 // lower bound on 64th-largest value
    __shared__ unsigned s_nc;                // candidate count

    const int b   = blockIdx.x;
    const int tid = threadIdx.x;

    const float t      = temps[b];
    const float safe_t = (t == 0.0f) ? 1.0f : t;

    const float* __restrict__ row  = logits     + (size_t)b * V;
    float*       __restrict__ srow = scaled_out + (size_t)b * V;

    const int V4 = V >> 2;
    const float4* __restrict__ row4  = (const float4*)row;
    float4*       __restrict__ srow4 = (float4*)srow;

    if (tid == 0) s_nc = 0u;

    // ---------------- pass 1: scale, store, per-thread max (HBM stream) ----
    float tmax = -INFINITY;

#if HAVE_ASYNC_COPY
    {
        const int CHUNK = TPB * 4;           // floats per chunk (8 KB)
        const int NC    = V / CHUNK;         // full chunks

        // Each wave's async op writes only its own lanes' 16B slots, and each
        // thread later reads exactly its own slot -> per-wave ASYNCcnt
        // ordering suffices; no __syncthreads in this loop.
        auto issue = [&](int c) {
            const float* gsrc = row + (size_t)c * CHUNK + (size_t)tid * 4;
            __builtin_amdgcn_global_load_async_to_lds_b128(
                (gv4i_p)(void*)gsrc,
                (lv4i_p)(void*)&s_abuf[c & (NBUF - 1)][tid],
                /*offset=*/0, /*cpol=*/0);
        };
        auto process = [&](int c) {
            float4 x = s_abuf[c & (NBUF - 1)][tid];
            x.x /= safe_t; x.y /= safe_t; x.z /= safe_t; x.w /= safe_t;
            srow4[(size_t)c * TPB + tid] = x;
            tmax = fmaxf(tmax, fmaxf(fmaxf(x.x, x.y), fmaxf(x.z, x.w)));
        };

        const int pre = (NC < NBUF) ? NC : NBUF;
        for (int k = 0; k < pre; ++k) issue(k);

        int c = 0;
        const int main_end = (NC >= NBUF) ? (NC - (NBUF - 1)) : 0;
        for (; c < main_end; ++c) {          // steady state: 4 chunks in flight
            WAIT_ASYNC(NBUF - 1);            // oldest chunk retired
            COMPILER_MEM_BARRIER();
            process(c);
            if (c + NBUF < NC) issue(c + NBUF);
        }
        for (; c < NC; ++c) {                // drain
            WAIT_ASYNC(0);
            COMPILER_MEM_BARRIER();
            process(c);
        }
        // tail (V not a multiple of CHUNK)
        for (int i = NC * CHUNK + tid; i < V; i += TPB) {
            float x = row[i] / safe_t;
            srow[i] = x;
            tmax = fmaxf(tmax, x);
        }
    }
#else
    for (int i = tid; i < V4; i += TPB) {
        float4 x = row4[i];
        x.x /= safe_t; x.y /= safe_t; x.z /= safe_t; x.w /= safe_t;
        srow4[i] = x;
        tmax = fmaxf(tmax, fmaxf(fmaxf(x.x, x.y), fmaxf(x.z, x.w)));
    }
    for (int i = (V4 << 2) + tid; i < V; i += TPB) {
        float x = row[i] / safe_t;
        srow[i] = x;
        tmax = fmaxf(tmax, x);
    }
#endif

    s_red[tid] = tmax;
    __syncthreads();

    // Exact rank of this thread's max among the TPB maxima (descending,
    // ties -> lower tid first): strict total order, ranks are a permutation.
    // rank 0 = global max m; rank 63 = q, a guaranteed lower bound on the
    // 64th-largest *element* of the row (the top-64 thread maxima are 64
    // distinct elements, all >= q).
    int r = 0;
    for (int j = 0; j < TPB; ++j) {
        float vj = s_red[j];
        r += (vj > tmax || (vj == tmax && j < tid)) ? 1 : 0;
    }
    if (r == 0)        s_m = tmax;
    if (r == KMAX - 1) s_q = tmax;
    __syncthreads();
    const float m = s_m;
    const float q = s_q;

    // ---------------- pass 2: softmax denom + candidate filter (L2 stream) --
    // Recompute x = logits/safe_t (IEEE div -> bitwise identical to pass 1);
    // avoids any global store->load hazard on srow.
    float sum = 0.0f;
    for (int i = tid; i < V4; i += TPB) {
        if (i + 8 * TPB < V4)
            __builtin_prefetch(&row4[i + 8 * TPB], 0, 3);   // global_prefetch_b8
        float4 x = row4[i];
        x.x /= safe_t; x.y /= safe_t; x.z /= safe_t; x.w /= safe_t;
        sum += expf(x.x - m) + expf(x.y - m) + expf(x.z - m) + expf(x.w - m);
        const int base = i << 2;
        #define PUSH_CAND(val, idx)                                   \
            if ((val) >= q) {                                         \
                unsigned p = atomicAdd(&s_nc, 1u);                    \
                if (p < CAP) { s_cv[p] = (val); s_ci[p] = (idx); }    \
            }
        PUSH_CAND(x.x, base + 0)
        PUSH_CAND(x.y, base + 1)
        PUSH_CAND(x.z, base + 2)
        PUSH_CAND(x.w, base + 3)
    }
    for (int i = (V4 << 2) + tid; i < V; i += TPB) {
        float x = row[i] / safe_t;
        sum += expf(x - m);
        PUSH_CAND(x, i)
    }
    #undef PUSH_CAND

    __syncthreads();                 // candidates + s_nc visible; s_red reads done
    s_red[tid] = sum;
    __syncthreads();
    for (int off = TPB >> 1; off > 0; off >>= 1) {
        if (tid < off) s_red[tid] += s_red[tid + off];
        __syncthreads();
    }
    const float denom = s_red[0];
    const int nc = (int)((s_nc < (unsigned)CAP) ? s_nc : (unsigned)CAP);

    // ---------------- exact stable rank-sort of candidates, keep top-64 -----
    // comparator: larger value first; tie -> lower original index first
    // (matches jnp.argsort(-probs) stable-sort semantics). Order of the
    // candidate buffer (atomic race) does not affect the result.
    for (int c = tid; c < nc; c += TPB) {
        const float v   = s_cv[c];
        const int   idx = s_ci[c];
        int rank = 0;
        for (int j = 0; j < nc; ++j) {
            float vj = s_cv[j];
            rank += (vj > v || (vj == v && s_ci[j] < idx)) ? 1 : 0;
        }
        if (rank < KMAX) { s_sv[rank] = v; s_si[rank] = idx; }
    }
    __syncthreads();

    // ---------------- final sample: joint top-k/top-p + Gumbel-max ----------
    if (tid == 0) {
        int token;
        if (t == 0.0f) {
            token = s_si[0];                    // greedy argmax(logits)
        } else {
            const float topp = top_ps[b];
            const float topk = top_ks[b];
            const float* __restrict__ nrow = noise + (size_t)b * V;
            const int K = (nc < KMAX) ? nc : KMAX;   // nc >= 64 by construction
            float cdf  = 0.0f;                       // exclusive cumsum
            float best = -INFINITY;
            int   bj   = 0;
            for (int j = 0; j < K; ++j) {
                const float sp   = expf(s_sv[j] - m) / denom;
                const bool  keep = ((float)j < topk) && (cdf <= topp);
                cdf += sp;
                if (keep) {
                    // noise consumed at *sorted position* j, as in reference
                    const float g  = -logf(-logf(nrow[j]));
                    const float sc = logf(sp) + g;
                    if (sc > best) { best = sc; bj = j; }  // first-max wins
                }
            }
            token = s_si[bj];
        }
        tok_out[b] = token;
    }
}

extern "C" void kernel_launch(void* const* d_in, const int* in_sizes, int n_in,
                              void* d_out, int out_size, void* d_ws, size_t ws_size,
                              hipStream_t stream) {
    const float* logits = (const float*)d_in[0];
    const float* temps  = (const float*)d_in[1];
    const float* top_ps = (const float*)d_in[2];
    const float* top_ks = (const float*)d_in[3];
    const float* noise  = (const float*)d_in[4];

    const int B = in_sizes[1];              // temperatures: (B,1)
    const int V = in_sizes[0] / B;          // logits: (B,V)

    // tuple output, concatenated flat in return order:
    //   [0, B)            token ids (int32 bits)
    //   [B, B + B*V)      scaled logits (float32)
    int*   tok_out    = (int*)d_out;
    float* scaled_out = (float*)d_out + B;

    sampler_topk_topp_kernel<<<B, TPB, 0, stream>>>(
        logits, temps, top_ps, top_ks, noise, tok_out, scaled_out, V);
}